// LinearDynamicalTF_2465311228008
// MI455X (gfx1250) — compile-verified
//
#include <hip/hip_runtime.h>

typedef float v2f __attribute__((ext_vector_type(2)));
typedef float v8f __attribute__((ext_vector_type(8)));

#define BB    64
#define TT    16384
#define CINX  8
#define COUTX 8
#define NTAP  4          // N_B + 1
#define KD    32         // NTAP * CINX
#define CH    512        // chunk length for the AR scan
#define NCH   (TT / CH)  // 32 chunks
#define TPB   256

// ---------------------------------------------------------------------------
// K1: FIR part as fp32 WMMA GEMM with LDS staging.
// bu[t, o] = sum_{k=0..31} U[t, k] * Bc[k, o],  k = tap*8 + cin,
// U[t, tap*8+cin] = u[b, t-3+tap, cin] (zero for t-3+tap < 0),
// Bc[k, o] = b_coeff[tap, cin, o], columns o>=8 zero-padded.
//
// Block = 256 threads = 8 waves; block covers 128 timesteps of one batch.
// Stage u rows [TB0-3, TB0+127] and padded b_coeff into LDS (coalesced b128),
// each wave computes one 16-timestep tile with 8 accumulating
// V_WMMA_F32_16X16X4_F32 (A frags are aligned ds_load_b64, B frags branchless),
// then results bounce through LDS for fully coalesced b128 global stores.
// ---------------------------------------------------------------------------
__global__ __launch_bounds__(TPB) void k_fir_wmma(const float* __restrict__ u,
                                                  const float* __restrict__ bc,
                                                  float* __restrict__ out) {
  __shared__ float su[131 * CINX + 8];  // 1048 used: rows TB0-3 .. TB0+127
  __shared__ float sb[KD * 16];         // b_coeff padded to [32][16]
  __shared__ float so[8 * 128];         // per-wave 16x8 output staging

  const int tid  = threadIdx.x;
  const int lane = tid & 31;
  const int wave = tid >> 5;
  const int b    = blockIdx.x >> 7;             // TT/128 = 128 blocks per batch
  const int TB0  = (blockIdx.x & 127) << 7;     // first timestep of block

  // ---- stage u (coalesced float4; negative-time guard only for block 0) ----
  const float* ub = u + (size_t)b * TT * CINX;
  for (int e4 = tid; e4 < 262; e4 += TPB) {     // 262*4 = 1048 floats
    const int g = (TB0 - 3) * CINX + e4 * 4;    // 16B-aligned when g >= 0
    if (g >= 0) {
      *(float4*)&su[e4 * 4] = *(const float4*)&ub[g];
    } else {
#pragma unroll
      for (int q = 0; q < 4; ++q)
        su[e4 * 4 + q] = (g + q >= 0) ? ub[g + q] : 0.0f;
    }
  }
  // ---- stage b_coeff zero-padded to 16 columns ----
  for (int e = tid; e < KD * 16; e += TPB) {
    const int k = e >> 4, n = e & 15;
    sb[e] = (n < COUTX) ? bc[k * COUTX + n] : 0.0f;
  }
  __syncthreads();

  // ---- WMMA fragments per ISA layout ----
  const int m    = lane & 15;   // A row (time within tile); also B/D column n
  const int half = lane >> 4;   // 0 -> K rows {0,1}, 1 -> K rows {2,3}
  const int kr   = half * 2;

  v8f acc = {};
#pragma unroll
  for (int g = 0; g < 8; ++g) {            // K = 32 in steps of 4
    const int kg  = 4 * g + kr;            // even -> (kg, kg+1) same tap
    const int tap = kg >> 3;
    const int ci  = kg & 7;                // even channel index
    // A: u rows are at LDS row (wave*16 + m + tap); 8-byte aligned pair.
    const v2f af = *(const v2f*)&su[(wave * 16 + m + tap) * CINX + ci];
    v2f bf;
    bf[0] = sb[kg * 16 + m];
    bf[1] = sb[(kg + 1) * 16 + m];
    acc = __builtin_amdgcn_wmma_f32_16x16x4_f32(false, af, false, bf,
                                                (short)0, acc, false, false);
  }

  // ---- stage D tile to LDS: VGPR r -> (time = half*8 + r, channel = m) ----
  if (m < COUTX) {
#pragma unroll
    for (int r = 0; r < 8; ++r)
      so[wave * 128 + (half * 8 + r) * COUTX + m] = acc[r];
  }
  __syncthreads();

  // ---- coalesced b128 store: wave's tile is 128 contiguous floats in out ----
  const float4 vo = *(const float4*)&so[wave * 128 + lane * 4];
  *(float4*)&out[((size_t)b * TT + TB0 + wave * 16) * COUTX + lane * 4] = vo;
}

// ---------------------------------------------------------------------------
// K2: per-(b, o, chunk) particular solution with zero incoming state.
// In-place over d_out; stores p[L-1], p[L-2], p[L-3] to workspace.
// ---------------------------------------------------------------------------
__global__ __launch_bounds__(TPB) void k_part(const float* __restrict__ acoef,
                                              float* __restrict__ out,
                                              float* __restrict__ pend) {
  const int tid = blockIdx.x * TPB + threadIdx.x;   // BB*NCH*COUTX = 16384
  const int o = tid & 7;
  const int c = (tid >> 3) & (NCH - 1);
  const int b = tid >> 8;                           // NCH*COUTX = 256

  const float a0 = acoef[0 * COUTX + o];
  const float a1 = acoef[1 * COUTX + o];
  const float a2 = acoef[2 * COUTX + o];

  float x1 = 0.f, x2 = 0.f, x3 = 0.f;
  float* p = out + ((size_t)b * TT + c * CH) * COUTX + o;
  for (int j = 0; j < CH; ++j) {
    if ((j & 3) == 0 && j + 32 < CH)
      __builtin_prefetch(p + (j + 32) * COUTX);     // global_prefetch_b8
    float x = p[j * COUTX] + a0 * x1 + a1 * x2 + a2 * x3;
    p[j * COUTX] = x;
    x3 = x2; x2 = x1; x1 = x;
  }
  float* pe = pend + ((size_t)((b * NCH + c) * COUTX + o)) * 3;
  pe[0] = x1; pe[1] = x2; pe[2] = x3;
}

// ---------------------------------------------------------------------------
// K3: homogeneous basis responses. c_k(j) = a0 c_k(j-1)+a1 c_k(j-2)+a2 c_k(j-3)
// with c_k(-1-i) = delta(i,k).  x_hom[j] = c1(j)x[-1] + c2(j)x[-2] + c3(j)x[-3].
// ---------------------------------------------------------------------------
__global__ void k_coeff(const float* __restrict__ acoef,
                        float* __restrict__ wc) {
  const int tid = threadIdx.x;
  if (tid >= 24) return;
  const int k = tid >> 3;   // basis 0..2
  const int o = tid & 7;
  const float a0 = acoef[o], a1 = acoef[COUTX + o], a2 = acoef[2 * COUTX + o];
  float p1 = (k == 0) ? 1.f : 0.f;
  float p2 = (k == 1) ? 1.f : 0.f;
  float p3 = (k == 2) ? 1.f : 0.f;
  float* w = wc + (size_t)(k * COUTX + o) * CH;
  for (int j = 0; j < CH; ++j) {
    float cc = a0 * p1 + a1 * p2 + a2 * p3;
    w[j] = cc;
    p3 = p2; p2 = p1; p1 = cc;
  }
}

// ---------------------------------------------------------------------------
// K4: chain chunk states sequentially (32 affine steps per (b,o)).
// s = (x[t0-1], x[t0-2], x[t0-3]); chunk 0 state comes from x0.
// ---------------------------------------------------------------------------
__global__ __launch_bounds__(TPB) void k_chain(const float* __restrict__ x0,
                                               const float* __restrict__ wc,
                                               const float* __restrict__ pend,
                                               float* __restrict__ st) {
  const int tid = blockIdx.x * TPB + threadIdx.x;
  if (tid >= BB * COUTX) return;
  const int o = tid & 7;
  const int b = tid >> 3;

  float s1 = x0[((size_t)b * 3 + 2) * COUTX + o];   // x[-1]
  float s2 = x0[((size_t)b * 3 + 1) * COUTX + o];   // x[-2]
  float s3 = x0[((size_t)b * 3 + 0) * COUTX + o];   // x[-3]

  const float* c1 = wc + (size_t)(0 * COUTX + o) * CH;
  const float* c2 = wc + (size_t)(1 * COUTX + o) * CH;
  const float* c3 = wc + (size_t)(2 * COUTX + o) * CH;
  const float c1a = c1[CH - 1], c1b = c1[CH - 2], c1c = c1[CH - 3];
  const float c2a = c2[CH - 1], c2b = c2[CH - 2], c2c = c2[CH - 3];
  const float c3a = c3[CH - 1], c3b = c3[CH - 2], c3c = c3[CH - 3];

  for (int c = 0; c < NCH; ++c) {
    float* s = st + ((size_t)((b * NCH + c) * COUTX + o)) * 3;
    s[0] = s1; s[1] = s2; s[2] = s3;
    const float* pe = pend + ((size_t)((b * NCH + c) * COUTX + o)) * 3;
    float n1 = pe[0] + c1a * s1 + c2a * s2 + c3a * s3;  // x[end]
    float n2 = pe[1] + c1b * s1 + c2b * s2 + c3b * s3;  // x[end-1]
    float n3 = pe[2] + c1c * s1 + c2c * s2 + c3c * s3;  // x[end-2]
    s1 = n1; s2 = n2; s3 = n3;
  }
}

// ---------------------------------------------------------------------------
// K5: fully parallel fix-up: x[t] = p[j] + c1(j)s1 + c2(j)s2 + c3(j)s3.
// ---------------------------------------------------------------------------
__global__ __launch_bounds__(TPB) void k_fix(const float* __restrict__ wc,
                                             const float* __restrict__ st,
                                             float* __restrict__ out) {
  const size_t idx = (size_t)blockIdx.x * TPB + threadIdx.x;  // BB*TT*COUTX
  const int o = (int)(idx & 7);
  const size_t bt = idx >> 3;
  const int t = (int)(bt & (TT - 1));
  const int b = (int)(bt >> 14);       // TT = 2^14
  const int c = t >> 9;                // CH = 512
  const int j = t & (CH - 1);
  const float* s = st + ((size_t)((b * NCH + c) * COUTX + o)) * 3;
  const float c1 = wc[(size_t)(0 * COUTX + o) * CH + j];
  const float c2 = wc[(size_t)(1 * COUTX + o) * CH + j];
  const float c3 = wc[(size_t)(2 * COUTX + o) * CH + j];
  out[idx] += c1 * s[0] + c2 * s[1] + c3 * s[2];
}

// ---------------------------------------------------------------------------
extern "C" void kernel_launch(void* const* d_in, const int* in_sizes, int n_in,
                              void* d_out, int out_size, void* d_ws, size_t ws_size,
                              hipStream_t stream) {
  (void)in_sizes; (void)n_in; (void)out_size; (void)ws_size;
  const float* u  = (const float*)d_in[0];   // [B, T, CIN]
  const float* x0 = (const float*)d_in[1];   // [B, 3, COUT]
  const float* ac = (const float*)d_in[2];   // [3, COUT]
  const float* bc = (const float*)d_in[3];   // [4, CIN, COUT]
  float* out = (float*)d_out;                // [B, T, COUT]

  float* ws   = (float*)d_ws;
  float* wc   = ws;                          // 3*8*CH        = 12288 floats
  float* pend = ws + 3 * COUTX * CH;         // B*NCH*8*3     = 49152 floats
  float* st   = pend + BB * NCH * COUTX * 3; // B*NCH*8*3     = 49152 floats

  // K1: FIR via fp32 WMMA with LDS staging. Block = 128 timesteps of a batch.
  k_fir_wmma<<<BB * (TT / 128), TPB, 0, stream>>>(u, bc, out);
  // K3 is independent of K1/K2; order on stream is fine.
  k_coeff<<<1, 32, 0, stream>>>(ac, wc);
  // K2: per-chunk particular scan (in place on out).
  k_part<<<(BB * NCH * COUTX) / TPB, TPB, 0, stream>>>(ac, out, pend);
  // K4: chain the 32 chunk states per (b, o).
  k_chain<<<2, TPB, 0, stream>>>(x0, wc, pend, st);
  // K5: parallel homogeneous correction.
  k_fix<<<(BB * TT * COUTX) / TPB, TPB, 0, stream>>>(wc, st, out);
}